// GCN_62242666053924
// MI455X (gfx1250) — compile-verified
//
#include <hip/hip_runtime.h>
#include <math.h>

typedef __attribute__((ext_vector_type(2))) float v2f;
typedef __attribute__((ext_vector_type(8))) float v8f;

#define D 128
#define TPB 256

// ---------------------------------------------------------------------------
// Degree / normalization
// ---------------------------------------------------------------------------
__global__ void deg_init_kernel(float* __restrict__ deg, int n) {
    int i = blockIdx.x * TPB + threadIdx.x;
    if (i < n) deg[i] = 1.0f;  // self-loop contributes 1 to every node's degree
}

__global__ void deg_accum_kernel(const int* __restrict__ dst, float* __restrict__ deg, int e) {
    int i = blockIdx.x * TPB + threadIdx.x;
    if (i < e) atomicAdd(&deg[dst[i]], 1.0f);
}

__global__ void make_dis_kernel(float* __restrict__ deg, int n) {
    int i = blockIdx.x * TPB + threadIdx.x;
    if (i < n) {
        float d = deg[i];
        deg[i] = (d > 0.0f) ? rsqrtf(d) : 0.0f;  // becomes dis[]
    }
}

// ---------------------------------------------------------------------------
// H = X @ W  via V_WMMA_F32_16X16X4_F32.
// grid.x = M/16 row tiles; block = 256 = 8 waves; wave w owns cols [16w,16w+16).
// A strip (16x128) staged in LDS once per block; EXEC is all-1s (no divergence).
// ---------------------------------------------------------------------------
__global__ void __launch_bounds__(TPB)
gemm_wmma_f32(const float* __restrict__ X, const float* __restrict__ W,
              float* __restrict__ H) {
    __shared__ float As[16 * D];  // 8 KB

    const int tid     = threadIdx.x;
    const int rowBase = blockIdx.x * 16;

    // Cooperative load of the 16x128 A strip.
    for (int i = tid; i < 16 * D; i += TPB) {
        As[i] = X[(size_t)rowBase * D + i];
    }
    __syncthreads();

    const int wave    = tid >> 5;
    const int lane    = tid & 31;
    const int colBase = wave * 16;
    const int rc      = lane & 15;        // A row / B col for this lane
    const int khalf   = (lane >> 4) * 2;  // 0 or 2

    v8f acc = {};
#pragma unroll 4
    for (int k = 0; k < D; k += 4) {
        const int ka = k + khalf;
        v2f a, b;
        a.x = As[rc * D + ka];
        a.y = As[rc * D + ka + 1];
        b.x = W[(size_t)ka * D + colBase + rc];
        b.y = W[(size_t)(ka + 1) * D + colBase + rc];
        acc = __builtin_amdgcn_wmma_f32_16x16x4_f32(
            /*neg_a=*/false, a, /*neg_b=*/false, b,
            /*c_mod=*/(short)0, acc, /*reuse_a=*/false, /*reuse_b=*/false);
    }

    // C/D layout: VGPR v -> M = v + 8*(lane>=16), N = lane&15
    const int n    = lane & 15;
    const int mofs = (lane >> 4) * 8;
#pragma unroll
    for (int v = 0; v < 8; ++v) {
        H[(size_t)(rowBase + mofs + v) * D + colBase + n] = acc[v];
    }
}

// ---------------------------------------------------------------------------
// out[i][:] = b[:] + dis[i]^2 * h[i][:]   (self-loop term + bias)
// float4 per thread; N*D/4 quads.
// ---------------------------------------------------------------------------
__global__ void self_bias_init_kernel(const float* __restrict__ h,
                                      const float* __restrict__ dis,
                                      const float* __restrict__ b,
                                      float* __restrict__ out, int n) {
    int idx = blockIdx.x * TPB + threadIdx.x;  // quad index
    int i = idx >> 5;                          // node (32 quads per node)
    int q = idx & 31;
    if (i >= n) return;
    float s = dis[i];
    s = s * s;
    float4 hv = ((const float4*)(h + (size_t)i * D))[q];
    float4 bv = ((const float4*)b)[q];
    float4 ov;
    ov.x = bv.x + hv.x * s;
    ov.y = bv.y + hv.y * s;
    ov.z = bv.z + hv.z * s;
    ov.w = bv.w + hv.w * s;
    ((float4*)(out + (size_t)i * D))[q] = ov;
}

// ---------------------------------------------------------------------------
// Edge aggregation: one wave32 per edge, each lane handles a float4 chunk.
// out[dst] += dis[src]*dis[dst] * h[src]
// ---------------------------------------------------------------------------
__global__ void edge_agg_kernel(const float* __restrict__ h,
                                const int* __restrict__ src,
                                const int* __restrict__ dst,
                                const float* __restrict__ dis,
                                float* __restrict__ out, int e) {
    int t = blockIdx.x * TPB + threadIdx.x;
    int ei = t >> 5;
    if (ei >= e) return;
    int lane = t & 31;
    int s = src[ei];
    int d = dst[ei];
    float w = dis[s] * dis[d];
    float4 hv = ((const float4*)(h + (size_t)s * D))[lane];
    float* o = out + (size_t)d * D + lane * 4;
    atomicAdd(o + 0, hv.x * w);
    atomicAdd(o + 1, hv.y * w);
    atomicAdd(o + 2, hv.z * w);
    atomicAdd(o + 3, hv.w * w);
}

// ---------------------------------------------------------------------------
// In-place ReLU over n4 float4 quads.
// ---------------------------------------------------------------------------
__global__ void relu_kernel(float* __restrict__ a, int n4) {
    int i = blockIdx.x * TPB + threadIdx.x;
    if (i >= n4) return;
    float4 v = ((float4*)a)[i];
    v.x = fmaxf(v.x, 0.0f);
    v.y = fmaxf(v.y, 0.0f);
    v.z = fmaxf(v.z, 0.0f);
    v.w = fmaxf(v.w, 0.0f);
    ((float4*)a)[i] = v;
}

// ---------------------------------------------------------------------------
// out = log_softmax(relu(in)) per row of 128; one wave32 per row,
// 4 strided features per lane; wave reductions via __shfl_xor (wave32).
// Safe in place: each wave touches only its own row.
// ---------------------------------------------------------------------------
__global__ void relu_logsoftmax_kernel(const float* __restrict__ in,
                                       float* __restrict__ out, int n) {
    int t = blockIdx.x * TPB + threadIdx.x;
    int row = t >> 5;
    if (row >= n) return;
    int lane = t & 31;
    const float* r = in + (size_t)row * D;

    float v[4];
    float m = -INFINITY;
#pragma unroll
    for (int k = 0; k < 4; ++k) {
        v[k] = fmaxf(r[lane + k * 32], 0.0f);  // relu
        m = fmaxf(m, v[k]);
    }
#pragma unroll
    for (int off = 16; off > 0; off >>= 1) m = fmaxf(m, __shfl_xor(m, off, 32));

    float s = 0.0f;
#pragma unroll
    for (int k = 0; k < 4; ++k) s += __expf(v[k] - m);
#pragma unroll
    for (int off = 16; off > 0; off >>= 1) s += __shfl_xor(s, off, 32);

    float ls = __logf(s);
    float* o = out + (size_t)row * D;
#pragma unroll
    for (int k = 0; k < 4; ++k) o[lane + k * 32] = v[k] - m - ls;
}

// ---------------------------------------------------------------------------
// Launch
// ---------------------------------------------------------------------------
extern "C" void kernel_launch(void* const* d_in, const int* in_sizes, int n_in,
                              void* d_out, int out_size, void* d_ws, size_t ws_size,
                              hipStream_t stream) {
    const float* x  = (const float*)d_in[0];
    const int*   ei = (const int*)d_in[1];
    const float* W0 = (const float*)d_in[2];
    const float* W1 = (const float*)d_in[3];
    const float* W2 = (const float*)d_in[4];
    const float* b0 = (const float*)d_in[5];
    const float* b1 = (const float*)d_in[6];
    const float* b2 = (const float*)d_in[7];

    const int N = in_sizes[0] / D;      // 100000
    const int E = in_sizes[1] / 2;      // 1600000
    const int* src = ei;
    const int* dst = ei + E;

    // Workspace layout: dis[N] (padded) | H[N*D] | AGG[N*D]
    float* dis = (float*)d_ws;
    size_t npad = ((size_t)N + 255) & ~(size_t)255;
    float* H   = dis + npad;
    float* AGG = H + (size_t)N * D;
    float* OUT = (float*)d_out;

    const int gN    = (N + TPB - 1) / TPB;           // node-wise
    const int gE    = (E + TPB - 1) / TPB;           // edge-wise (1 thr/edge)
    const int gEW   = (E * 32 + TPB - 1) / TPB;      // wave per edge
    const int gQ    = ((size_t)N * 32 + TPB - 1) / TPB;  // float4 quads of N*D
    const int gRow  = (N * 32 + TPB - 1) / TPB;      // wave per row
    const int gGemm = N / 16;                        // 6250 row tiles (exact)

    // --- normalization: deg -> dis ---
    deg_init_kernel<<<gN, TPB, 0, stream>>>(dis, N);
    deg_accum_kernel<<<gE, TPB, 0, stream>>>(dst, dis, E);
    make_dis_kernel<<<gN, TPB, 0, stream>>>(dis, N);

    // --- layer 0: X -> AGG ---
    gemm_wmma_f32<<<gGemm, TPB, 0, stream>>>(x, W0, H);
    self_bias_init_kernel<<<gQ, TPB, 0, stream>>>(H, dis, b0, AGG, N);
    edge_agg_kernel<<<gEW, TPB, 0, stream>>>(H, src, dst, dis, AGG, E);
    relu_kernel<<<gQ, TPB, 0, stream>>>(AGG, N * 32);

    // --- layer 1: AGG -> AGG ---
    gemm_wmma_f32<<<gGemm, TPB, 0, stream>>>(AGG, W1, H);
    self_bias_init_kernel<<<gQ, TPB, 0, stream>>>(H, dis, b1, AGG, N);
    edge_agg_kernel<<<gEW, TPB, 0, stream>>>(H, src, dst, dis, AGG, E);
    relu_kernel<<<gQ, TPB, 0, stream>>>(AGG, N * 32);

    // --- layer 2: AGG -> OUT, then fused relu+log_softmax in place ---
    gemm_wmma_f32<<<gGemm, TPB, 0, stream>>>(AGG, W2, H);
    self_bias_init_kernel<<<gQ, TPB, 0, stream>>>(H, dis, b2, OUT, N);
    edge_agg_kernel<<<gEW, TPB, 0, stream>>>(H, src, dst, dis, OUT, E);
    relu_logsoftmax_kernel<<<gRow, TPB, 0, stream>>>(OUT, OUT, N);
}